// KalmanPredictorSimpleMambaV1_46591805227532
// MI455X (gfx1250) — compile-verified
//
#include <hip/hip_runtime.h>
#include <hip/hip_bf16.h>

// ---------------- problem constants ----------------
#define BB   2
#define CH   32
#define NL   2
#define DE   384
#define DI   768
#define NST  8
#define RK   64
#define GRP  64
#define HH   48
#define WW   48
#define LL   (HH*WW)        // 2304
#define CPJ  (RK + 2*NST)   // 80

typedef __attribute__((ext_vector_type(2))) float v2f;
typedef __attribute__((ext_vector_type(8))) float v8f;
typedef __attribute__((ext_vector_type(4))) unsigned int u32x4;
typedef __attribute__((ext_vector_type(4))) int i32x4;
typedef __attribute__((ext_vector_type(8))) int i32x8;

// ---- TDM availability / arity probing (compile-time) ----
#if defined(__has_builtin)
# if __has_builtin(__builtin_amdgcn_tensor_load_to_lds)
#  define HAVE_TDM 1
# endif
#endif
#ifndef HAVE_TDM
# define HAVE_TDM 0
#endif
#if __has_include(<hip/amd_detail/amd_gfx1250_TDM.h>)
# define TDM_6ARG 1
#else
# define TDM_6ARG 0
#endif

// =====================================================================
// TDM: DMA a (rows x 64) f32 tile of row-major B (ld = N) into LDS.
// Builds the D# descriptor per CDNA5 ISA ch.8 (2-D tile, data_size=4B).
// =====================================================================
__device__ __forceinline__ void tdm_load_panel(unsigned lds_addr,
                                               const float* gptr,
                                               int N, int Kdim,
                                               int rows, int cols)
{
#if HAVE_TDM
    unsigned long long ga = (unsigned long long)gptr;
    u32x4 g0;
    g0[0] = 1u;                                              // count=1, user mode
    g0[1] = lds_addr;                                        // LDS byte address
    g0[2] = (unsigned)ga;                                    // global_addr[31:0]
    g0[3] = ((unsigned)(ga >> 32) & 0x01FFFFFFu) | (2u << 30); // addr[56:32] | type=2
    i32x8 g1;
    g1[0] = (int)(2u << 16);                                 // data_size = 4 bytes
    g1[1] = (int)(((unsigned)N & 0xFFFFu) << 16);            // tensor_dim0[15:0] @63:48
    g1[2] = (int)((((unsigned)N >> 16) & 0xFFFFu) |
                  (((unsigned)Kdim & 0xFFFFu) << 16));       // dim0 hi | tensor_dim1 lo
    g1[3] = (int)((((unsigned)Kdim >> 16) & 0xFFFFu) |
                  (((unsigned)cols & 0xFFFFu) << 16));       // dim1 hi | tile_dim0
    g1[4] = (int)((unsigned)rows & 0xFFFFu);                 // tile_dim1 (tile_dim2=0)
    g1[5] = N;                                               // tensor_dim0_stride[31:0]
    g1[6] = 0;                                               // stride hi / dim1_stride lo
    g1[7] = 0;
    i32x4 zz = {0, 0, 0, 0};
  #if TDM_6ARG
    i32x8 zz8 = {0, 0, 0, 0, 0, 0, 0, 0};
    __builtin_amdgcn_tensor_load_to_lds(g0, g1, zz, zz, zz8, 0);
  #else
    __builtin_amdgcn_tensor_load_to_lds(g0, g1, zz, zz, 0);
  #endif
#else
    (void)lds_addr; (void)gptr; (void)N; (void)Kdim; (void)rows; (void)cols;
#endif
}

// =====================================================================
// LDS/TDM-staged WMMA f32 GEMM.
// Block = 256 thr (8 waves). Block tile: 128 (M) x 64 (N).
// Each wave: 16x64 strip = 4 accumulator tiles; A from global (L2-hot),
// B panel (16 x 64) DMA'd to LDS by TDM, double-buffered.
// Requires: M % 128 == 0, N % 64 == 0, K % 16 == 0.
// =====================================================================
__global__ __launch_bounds__(256)
void wmma_gemm_lds(const float* __restrict__ A,
                   const float* __restrict__ Bm,
                   float* __restrict__ C,
                   const float* __restrict__ bias,
                   int M, int N, int K,
                   long aBatchStride, int aMod,
                   long bBatchStride, long cBatchStride,
                   long biasStride, int biasMod,
                   int act)
{
    __shared__ float sB[2][16 * 64];

    const int wave  = threadIdx.x >> 5;
    const int lane  = threadIdx.x & 31;
    const int lhalf = lane >> 4;
    const int l16   = lane & 15;

    const int n0 = blockIdx.x * 64;
    const int tm = blockIdx.y * 128 + wave * 16;
    const int z  = blockIdx.z;

    const float* Ab = A + (long)(z % aMod) * aBatchStride
                        + (long)(tm + l16) * K + 2 * lhalf;
    const float* Bb = Bm + (long)z * bBatchStride + n0;
    float*       Cb = C + (long)z * cBatchStride;

    v8f zero8 = {};
    v8f acc[4];
    #pragma unroll
    for (int i = 0; i < 4; ++i) acc[i] = zero8;

    const unsigned ldsBase = (unsigned)(unsigned long long)(void*)&sB[0][0];
    const int P = K / 16;

#if HAVE_TDM
    if (wave == 0) {
        tdm_load_panel(ldsBase, Bb, N, K, 16, 64);
        __builtin_amdgcn_s_wait_tensorcnt((short)0);
    }
    __syncthreads();
#endif

    for (int p = 0; p < P; ++p) {
        float* sb = sB[p & 1];
#if HAVE_TDM
        if (wave == 0 && p + 1 < P)
            tdm_load_panel(ldsBase + (unsigned)((p + 1) & 1) * 16 * 64 * 4,
                           Bb + (long)(p + 1) * 16 * N, N, K, 16, 64);
#else
        __syncthreads();
        {   // cooperative staging fallback: 1024 floats / 256 threads
            int t = threadIdx.x;
            int r = (t * 4) / 64, c = (t * 4) % 64;
            const float4 v = *(const float4*)(Bb + (long)(p * 16 + r) * N + c);
            *(float4*)&sb[t * 4] = v;
        }
        __syncthreads();
#endif
        #pragma unroll
        for (int kk = 0; kk < 16; kk += 4) {
            const float* ap = Ab + (p * 16 + kk);
            v2f a; a.x = ap[0]; a.y = ap[1];
            const int r = kk + 2 * lhalf;
            #pragma unroll
            for (int ni = 0; ni < 4; ++ni) {
                v2f b;
                b.x = sb[r * 64 + ni * 16 + l16];
                b.y = sb[(r + 1) * 64 + ni * 16 + l16];
                acc[ni] = __builtin_amdgcn_wmma_f32_16x16x4_f32(
                              false, a, false, b, (short)0, acc[ni], false, false);
            }
        }
#if HAVE_TDM
        if (wave == 0 && p + 1 < P)
            __builtin_amdgcn_s_wait_tensorcnt((short)0);
        __syncthreads();
#endif
    }

    const int m0 = tm + 8 * lhalf;
    #pragma unroll
    for (int ni = 0; ni < 4; ++ni) {
        const int n = n0 + ni * 16 + l16;
        #pragma unroll
        for (int v = 0; v < 8; ++v) {
            const int m = m0 + v;
            float r = acc[ni][v];
            if (bias) r += bias[(long)(z % biasMod) * biasStride + m];
            if (act == 1) r = (r > 20.f) ? r : logf(1.f + __expf(r));
            Cb[(long)m * N + n] = r;
        }
    }
}

// =====================================================================
// Direct-global WMMA GEMM, 16x64 strip per wave (for M=80 case).
// Requires: M % 16 == 0, N % 256 == 0 (4 waves x 64), K % 4 == 0.
// =====================================================================
__global__ __launch_bounds__(128)
void wmma_gemm_n4(const float* __restrict__ A,
                  const float* __restrict__ Bm,
                  float* __restrict__ C,
                  const float* __restrict__ bias,
                  int M, int N, int K,
                  long aBatchStride, int aMod,
                  long bBatchStride, long cBatchStride,
                  long biasStride, int biasMod,
                  int act)
{
    const int wave  = threadIdx.x >> 5;
    const int lane  = threadIdx.x & 31;
    const int lhalf = lane >> 4;
    const int l16   = lane & 15;

    const int n0 = (blockIdx.x * 4 + wave) * 64;
    const int tm = blockIdx.y * 16;
    const int z  = blockIdx.z;

    const float* Ab = A + (long)(z % aMod) * aBatchStride
                        + (long)(tm + l16) * K + 2 * lhalf;
    const float* Bb = Bm + (long)z * bBatchStride;
    float*       Cb = C + (long)z * cBatchStride;

    v8f zero8 = {};
    v8f acc[4];
    #pragma unroll
    for (int i = 0; i < 4; ++i) acc[i] = zero8;

    #pragma unroll 2
    for (int k = 0; k < K; k += 4) {
        v2f a; a.x = Ab[k]; a.y = Ab[k + 1];
        const float* bp = Bb + (long)(k + 2 * lhalf) * N + n0 + l16;
        #pragma unroll
        for (int ni = 0; ni < 4; ++ni) {
            v2f b; b.x = bp[ni * 16]; b.y = bp[ni * 16 + N];
            acc[ni] = __builtin_amdgcn_wmma_f32_16x16x4_f32(
                          false, a, false, b, (short)0, acc[ni], false, false);
        }
    }

    const int m0 = tm + 8 * lhalf;
    #pragma unroll
    for (int ni = 0; ni < 4; ++ni) {
        const int n = n0 + ni * 16 + l16;
        #pragma unroll
        for (int v = 0; v < 8; ++v) {
            const int m = m0 + v;
            float r = acc[ni][v];
            if (bias) r += bias[(long)(z % biasMod) * biasStride + m];
            if (act == 1) r = (r > 20.f) ? r : logf(1.f + __expf(r));
            Cb[(long)m * N + n] = r;
        }
    }
}

// =====================================================================
// Dense 3x3 conv, SAME padding, NCHW / OIHW
// =====================================================================
__global__ void conv3x3_dense(const float* __restrict__ in,
                              const float* __restrict__ wgt,
                              const float* __restrict__ bias,
                              float* __restrict__ out,
                              int Bn, int CI, int CO)
{
    int idx = blockIdx.x * blockDim.x + threadIdx.x;
    int total = Bn * CO * LL;
    if (idx >= total) return;
    int w0 = idx % WW;
    int h0 = (idx / WW) % HH;
    int co = (idx / LL) % CO;
    int bb = idx / (LL * CO);

    float acc = bias[co];
    for (int ci = 0; ci < CI; ++ci) {
        const float* ip = in + ((long)bb * CI + ci) * LL;
        const float* wp = wgt + ((long)co * CI + ci) * 9;
        #pragma unroll
        for (int kh = 0; kh < 3; ++kh) {
            int hy = h0 + kh - 1;
            if ((unsigned)hy >= (unsigned)HH) continue;
            #pragma unroll
            for (int kw = 0; kw < 3; ++kw) {
                int wx = w0 + kw - 1;
                if ((unsigned)wx >= (unsigned)WW) continue;
                acc += ip[hy * WW + wx] * wp[kh * 3 + kw];
            }
        }
    }
    out[idx] = acc;
}

// =====================================================================
// GroupNorm: one block per (b, group); 6 channels * 2304 spatial
// =====================================================================
__global__ void groupnorm_kernel(const float* __restrict__ x,
                                 float* __restrict__ out,
                                 const float* __restrict__ w,
                                 const float* __restrict__ b)
{
    const int Cg = DE / GRP;                       // 6
    int g  = blockIdx.x % GRP;
    int bb = blockIdx.x / GRP;
    const float* xp = x + ((long)bb * DE + g * Cg) * LL;
    float* op = out + ((long)bb * DE + g * Cg) * LL;
    const int n = Cg * LL;

    float s = 0.f, s2 = 0.f;
    for (int i = threadIdx.x; i < n; i += blockDim.x) {
        float v = xp[i]; s += v; s2 += v * v;
    }
    __shared__ float r0[256], r1[256];
    r0[threadIdx.x] = s; r1[threadIdx.x] = s2;
    __syncthreads();
    for (int off = 128; off > 0; off >>= 1) {
        if (threadIdx.x < off) {
            r0[threadIdx.x] += r0[threadIdx.x + off];
            r1[threadIdx.x] += r1[threadIdx.x + off];
        }
        __syncthreads();
    }
    float mu  = r0[0] / n;
    float var = r1[0] / n - mu * mu;
    float inv = rsqrtf(var + 1e-5f);
    for (int i = threadIdx.x; i < n; i += blockDim.x) {
        int c = g * Cg + i / LL;
        op[i] = (xp[i] - mu) * inv * w[c] + b[c];
    }
}

// =====================================================================
// Depthwise 3x3 conv + SiLU on first DI rows of xz
// =====================================================================
__global__ void dwconv_silu(const float* __restrict__ xz,
                            const float* __restrict__ cw,
                            const float* __restrict__ cb,
                            float* __restrict__ xi)
{
    int idx = blockIdx.x * blockDim.x + threadIdx.x;
    int total = BB * DI * LL;
    if (idx >= total) return;
    int w0 = idx % WW;
    int h0 = (idx / WW) % HH;
    int d  = (idx / LL) % DI;
    int bb = idx / (LL * DI);

    const float* ip = xz + ((long)bb * (2 * DI) + d) * LL;
    const float* wp = cw + (long)d * 9;
    float acc = cb[d];
    #pragma unroll
    for (int kh = 0; kh < 3; ++kh) {
        int hy = h0 + kh - 1;
        if ((unsigned)hy >= (unsigned)HH) continue;
        #pragma unroll
        for (int kw = 0; kw < 3; ++kw) {
            int wx = w0 + kw - 1;
            if ((unsigned)wx >= (unsigned)WW) continue;
            acc += ip[hy * WW + wx] * wp[kh * 3 + kw];
        }
    }
    float sig = 1.f / (1.f + __expf(-acc));
    xi[idx] = acc * sig;
}

// =====================================================================
// Build the 4 scan directions xs[b,k,d,l]
// =====================================================================
__global__ void build_xs(const float* __restrict__ xi, float* __restrict__ xs)
{
    int idx = blockIdx.x * blockDim.x + threadIdx.x;
    int total = BB * DI * LL;
    if (idx >= total) return;
    int w0 = idx % WW;
    int h0 = (idx / WW) % HH;
    int d  = (idx / LL) % DI;
    int bb = idx / (LL * DI);

    float v = xi[idx];
    int l  = h0 * WW + w0;
    int lt = w0 * HH + h0;
    long kstride = (long)DI * LL;
    long base = ((long)(bb * 4) * DI + d) * LL;
    xs[base + l] = v;
    xs[base + kstride + lt] = v;
    xs[base + 2 * kstride + (LL - 1 - l)] = v;
    xs[base + 3 * kstride + (LL - 1 - lt)] = v;
}

// =====================================================================
// Repack Bs/Cs (rows 64..79 of x_dbl) into contiguous [z][l][16]
// =====================================================================
__global__ void repack_bc(const float* __restrict__ xdbl, float* __restrict__ bcpk)
{
    int idx = blockIdx.x * blockDim.x + threadIdx.x;
    int total = BB * 4 * LL;
    if (idx >= total) return;
    int l = idx % LL;
    int z = idx / LL;
    const float* src = xdbl + ((long)z * CPJ + RK) * LL + l;
    float* dst = bcpk + (long)idx * 16;
    #pragma unroll
    for (int n = 0; n < 16; ++n) dst[n] = src[(long)n * LL];
}

// =====================================================================
// Selective scan: one thread per (b,k,d) chain, sequential over L.
// oy written in-place over delta (read-before-write per thread).
// =====================================================================
__global__ void scan_kernel(const float* __restrict__ xs,
                            float* __restrict__ delta_oy,
                            const float* __restrict__ bcpk,
                            const float* __restrict__ alog,
                            const float* __restrict__ dsv)
{
    int z = blockIdx.z;                    // b*4 + k
    int k = z & 3;
    int d = blockIdx.x * blockDim.x + threadIdx.x;
    if (d >= DI) return;

    float Av[NST];
    #pragma unroll
    for (int n = 0; n < NST; ++n)
        Av[n] = -__expf(alog[((long)k * DI + d) * NST + n]);
    float Dv = dsv[k * DI + d];

    float h[NST];
    #pragma unroll
    for (int n = 0; n < NST; ++n) h[n] = 0.f;

    const float* up = xs + ((long)z * DI + d) * LL;
    float*       dp = delta_oy + ((long)z * DI + d) * LL;
    const float* bc = bcpk + (long)z * LL * 16;

    for (int l = 0; l < LL; ++l) {
        float u  = up[l];
        float dl = dp[l];
        float du = dl * u;
        const float* b8 = bc + l * 16;
        float y = 0.f;
        #pragma unroll
        for (int n = 0; n < NST; ++n) {
            h[n] = __expf(dl * Av[n]) * h[n] + du * b8[n];
            y += h[n] * b8[8 + n];
        }
        dp[l] = y + u * Dv;
    }
}

// =====================================================================
// Merge 4 scan directions back to [b,d,l]
// =====================================================================
__global__ void merge_kernel(const float* __restrict__ oy, float* __restrict__ y)
{
    int idx = blockIdx.x * blockDim.x + threadIdx.x;
    int total = BB * DI * LL;
    if (idx >= total) return;
    int w0 = idx % WW;
    int h0 = (idx / WW) % HH;
    int d  = (idx / LL) % DI;
    int bb = idx / (LL * DI);

    int l  = h0 * WW + w0;
    int lt = w0 * HH + h0;
    long kstride = (long)DI * LL;
    long base = ((long)(bb * 4) * DI + d) * LL;
    float v = oy[base + l]
            + oy[base + kstride + lt]
            + oy[base + 2 * kstride + (LL - 1 - l)]
            + oy[base + 3 * kstride + (LL - 1 - lt)];
    y[((long)bb * DI + d) * LL + l] = v;
}

// =====================================================================
// LayerNorm over d (per b,l) * scale/shift, then gate with silu(z)
// =====================================================================
__global__ void outnorm_gate(float* __restrict__ y,
                             const float* __restrict__ xz,
                             const float* __restrict__ onw,
                             const float* __restrict__ onb)
{
    int bl = blockIdx.x;
    int l  = bl % LL;
    int bb = bl / LL;
    float* yp = y + (long)bb * DI * LL + l;            // stride LL per d

    float s = 0.f, s2 = 0.f;
    for (int d = threadIdx.x; d < DI; d += blockDim.x) {
        float v = yp[(long)d * LL]; s += v; s2 += v * v;
    }
    __shared__ float r0[256], r1[256];
    r0[threadIdx.x] = s; r1[threadIdx.x] = s2;
    __syncthreads();
    for (int off = 128; off > 0; off >>= 1) {
        if (threadIdx.x < off) {
            r0[threadIdx.x] += r0[threadIdx.x + off];
            r1[threadIdx.x] += r1[threadIdx.x + off];
        }
        __syncthreads();
    }
    float mu  = r0[0] / DI;
    float var = r1[0] / DI - mu * mu;
    float inv = rsqrtf(var + 1e-5f);
    for (int d = threadIdx.x; d < DI; d += blockDim.x) {
        float v = (yp[(long)d * LL] - mu) * inv * onw[d] + onb[d];
        float zv = xz[((long)bb * (2 * DI) + DI + d) * LL + l];
        float sig = 1.f / (1.f + __expf(-zv));
        yp[(long)d * LL] = v * (zv * sig);
    }
}

// =====================================================================
// Host orchestration
// =====================================================================
extern "C" void kernel_launch(void* const* d_in, const int* in_sizes, int n_in,
                              void* d_out, int out_size, void* d_ws, size_t ws_size,
                              hipStream_t stream)
{
    const float* image    = (const float*)d_in[0];
    const float* expand_w = (const float*)d_in[1];
    const float* expand_b = (const float*)d_in[2];
    const float* shrink_w = (const float*)d_in[3];
    const float* shrink_b = (const float*)d_in[4];
    const float* gn_w     = (const float*)d_in[5];
    const float* gn_b     = (const float*)d_in[6];
    const float* in_w     = (const float*)d_in[7];
    const float* in_b     = (const float*)d_in[8];
    const float* conv_w   = (const float*)d_in[9];
    const float* conv_b   = (const float*)d_in[10];
    const float* xproj_w  = (const float*)d_in[11];
    const float* dtproj_w = (const float*)d_in[12];
    const float* dtproj_b = (const float*)d_in[13];
    const float* A_logs   = (const float*)d_in[14];
    const float* Ds_p     = (const float*)d_in[15];
    const float* onorm_w  = (const float*)d_in[16];
    const float* onorm_b  = (const float*)d_in[17];
    const float* out_w    = (const float*)d_in[18];
    const float* out_b    = (const float*)d_in[19];
    (void)in_sizes; (void)n_in; (void)out_size; (void)ws_size;

    // workspace carve-up (floats); OY aliases DELT, Y aliases XI
    float* X0   = (float*)d_ws;
    float* XN   = X0   + (size_t)BB * DE * LL;
    float* XZ   = XN   + (size_t)BB * DE * LL;
    float* XI   = XZ   + (size_t)BB * 2 * DI * LL;
    float* XS   = XI   + (size_t)BB * DI * LL;
    float* XDBL = XS   + (size_t)BB * 4 * DI * LL;
    float* BCPK = XDBL + (size_t)BB * 4 * CPJ * LL;
    float* DELT = BCPK + (size_t)BB * 4 * LL * 16;

    const int NB64 = LL / 64;               // 36 N-blocks for LDS-staged GEMM
    const int EW_BLK = 256;
    const int totBDEL = BB * DE * LL;
    const int totBDIL = BB * DI * LL;

    // expand conv: (B,32,H,W) -> (B,384,H,W)
    conv3x3_dense<<<(totBDEL + EW_BLK - 1) / EW_BLK, EW_BLK, 0, stream>>>(
        image, expand_w, expand_b, X0, BB, CH, DE);

    for (int l = 0; l < NL; ++l) {
        const float* gnw_l = gn_w + (size_t)l * DE;
        const float* gnb_l = gn_b + (size_t)l * DE;
        const float* inw_l = in_w + (size_t)l * 2 * DI * DE;
        const float* inb_l = in_b + (size_t)l * 2 * DI;
        const float* cw_l  = conv_w + (size_t)l * DI * 9;
        const float* cb_l  = conv_b + (size_t)l * DI;
        const float* xp_l  = xproj_w + (size_t)l * 4 * CPJ * DI;
        const float* dtw_l = dtproj_w + (size_t)l * 4 * DI * RK;
        const float* dtb_l = dtproj_b + (size_t)l * 4 * DI;
        const float* al_l  = A_logs + (size_t)l * 4 * DI * NST;
        const float* ds_l  = Ds_p   + (size_t)l * 4 * DI;
        const float* onw_l = onorm_w + (size_t)l * DI;
        const float* onb_l = onorm_b + (size_t)l * DI;
        const float* ow_l  = out_w + (size_t)l * DE * DI;
        const float* ob_l  = out_b + (size_t)l * DE;

        // group norm
        groupnorm_kernel<<<BB * GRP, 256, 0, stream>>>(X0, XN, gnw_l, gnb_l);

        // in_proj GEMM: (1536 x 384) x (384 x L) per b -> XZ  [TDM+LDS staged]
        wmma_gemm_lds<<<dim3(NB64, (2 * DI) / 128, BB), 256, 0, stream>>>(
            inw_l, XN, XZ, inb_l,
            2 * DI, LL, DE,
            0L, 1, (long)DE * LL, (long)2 * DI * LL,
            0L, 1, 0);

        // depthwise conv + silu -> XI
        dwconv_silu<<<(totBDIL + EW_BLK - 1) / EW_BLK, EW_BLK, 0, stream>>>(
            XZ, cw_l, cb_l, XI);

        // 4-direction scan inputs
        build_xs<<<(totBDIL + EW_BLK - 1) / EW_BLK, EW_BLK, 0, stream>>>(XI, XS);

        // x_proj GEMM: (80 x 768) x (768 x L) per (b,k) -> XDBL  [direct]
        wmma_gemm_n4<<<dim3(NB64 / 4, CPJ / 16, BB * 4), 128, 0, stream>>>(
            xp_l, XS, XDBL, nullptr,
            CPJ, LL, DI,
            (long)CPJ * DI, 4, (long)DI * LL, (long)CPJ * LL,
            0L, 1, 0);

        // dt_proj GEMM + softplus: (768 x 64) x (64 x L) per (b,k) -> DELT
        wmma_gemm_lds<<<dim3(NB64, DI / 128, BB * 4), 256, 0, stream>>>(
            dtw_l, XDBL, DELT, dtb_l,
            DI, LL, RK,
            (long)DI * RK, 4, (long)CPJ * LL, (long)DI * LL,
            (long)DI, 4, 1);

        // pack Bs/Cs contiguously for the scan
        repack_bc<<<(BB * 4 * LL + EW_BLK - 1) / EW_BLK, EW_BLK, 0, stream>>>(
            XDBL, BCPK);

        // selective scan (oy written over DELT)
        scan_kernel<<<dim3(DI / 256, 1, BB * 4), 256, 0, stream>>>(
            XS, DELT, BCPK, al_l, ds_l);

        // merge 4 directions -> Y (aliases XI)
        merge_kernel<<<(totBDIL + EW_BLK - 1) / EW_BLK, EW_BLK, 0, stream>>>(
            DELT, XI);

        // layer-norm over d + silu(z) gate
        outnorm_gate<<<BB * LL, 256, 0, stream>>>(XI, XZ, onw_l, onb_l);

        // out_proj GEMM: (384 x 768) x (768 x L) per b -> X0
        wmma_gemm_lds<<<dim3(NB64, DE / 128, BB), 256, 0, stream>>>(
            ow_l, XI, X0, ob_l,
            DE, LL, DI,
            0L, 1, (long)DI * LL, (long)DE * LL,
            0L, 1, 0);
    }

    // shrink conv: (B,384,H,W) -> (B,32,H,W)
    conv3x3_dense<<<(BB * CH * LL + EW_BLK - 1) / EW_BLK, EW_BLK, 0, stream>>>(
        X0, shrink_w, shrink_b, (float*)d_out, BB, DE, CH);
}